// PersonalizedFrontend_22213570854943
// MI455X (gfx1250) — compile-verified
//
#include <hip/hip_runtime.h>
#include <stdint.h>

#define USE_ASYNC_LDS 1

typedef __attribute__((ext_vector_type(2))) float v2f;
typedef __attribute__((ext_vector_type(4))) float v4f;
typedef __attribute__((ext_vector_type(8))) float v8f;

#define BATCH 32768
#define IN    1024
#define HID   30
#define NPAD  32
#define ODIM  8
#define NSUB  64

// ---------------------------------------------------------------------------
// Kernel A: h = relu(features @ bW0 + bb0), written padded as [BATCH][32].
// 256 WGs x 256 threads (8 waves). Each WG owns 128 rows; each wave a 16x32
// output tile via V_WMMA_F32_16X16X4_F32. Features double-buffered into LDS
// with global_load_async_to_lds_b128; bW0 staged transposed+padded once.
// Dynamic LDS: 32*1024 (BT) + 2*128*32 (feat) floats = 160KB.
// ---------------------------------------------------------------------------

__device__ __forceinline__ void stage_chunk(const float* __restrict__ gbase,
                                            float* lbase, int tid) {
#pragma unroll
  for (int i = 0; i < 4; ++i) {
    int q  = tid + i * 256;     // 0..1023 float4-slots: 128 rows x 8 slots
    int r  = q >> 3;            // row within chunk
    int kq = (q & 7) << 2;      // k offset within 32-wide chunk
    const float* g = gbase + (size_t)r * IN + kq;
    float* l = lbase + r * NPAD + kq;
#if USE_ASYNC_LDS
    unsigned laddr = (unsigned)(uintptr_t)l;             // LDS byte offset (low 32 bits)
    unsigned long long ga = (unsigned long long)(uintptr_t)g;
    asm volatile("global_load_async_to_lds_b128 %0, %1, off"
                 :: "v"(laddr), "v"(ga) : "memory");
#else
    *(v4f*)l = *(const v4f*)g;
#endif
  }
}

__global__ __launch_bounds__(256) void backend_gemm_kernel(
    const float* __restrict__ feat, const float* __restrict__ bW0,
    const float* __restrict__ bb0, float* __restrict__ hout) {
  extern __shared__ float smem[];
  float* sBT = smem;              // [32][1024]  bW0 transposed, cols 30/31 zero
  float* sF  = smem + NPAD * IN;  // 2 x [128][32] feature chunks
  const int tid  = threadIdx.x;
  const int row0 = blockIdx.x * 128;

  // Issue async stage of chunk 0 first so it overlaps the BT staging below.
  stage_chunk(feat + (size_t)row0 * IN, sF, tid);

  // Stage bW0 transposed + zero-padded: sBT[n*1024 + k]
  for (int idx = tid; idx < NPAD * IN; idx += 256) {
    int n = idx >> 10;
    int k = idx & (IN - 1);
    sBT[idx] = (n < HID) ? bW0[(size_t)k * HID + n] : 0.0f;
  }

  const int lane  = tid & 31;
  const int lanem = lane & 15;
  const int hi    = lane >> 4;   // 0: K{0,1}, 1: K{2,3} within a k-step
  const int wv    = tid >> 5;
  const int koff  = hi * 2;

  v8f acc0 = {0.f, 0.f, 0.f, 0.f, 0.f, 0.f, 0.f, 0.f};
  v8f acc1 = acc0;

  const float* fbase  = sF + (wv * 16 + lanem) * NPAD;  // A row for this lane
  const float* b0base = sBT + (size_t)lanem * IN;       // B col tile 0 (n = lanem)
  const float* b1base = sBT + (size_t)(16 + lanem) * IN;// B col tile 1 (n = 16+lanem)

  for (int kc = 0; kc < IN / 32; ++kc) {
#if USE_ASYNC_LDS
    asm volatile("s_wait_asynccnt 0x0" ::: "memory");   // my chunk-kc copies done
#endif
    __syncthreads();                                    // all waves' data visible
    if (kc + 1 < IN / 32)
      stage_chunk(feat + (size_t)row0 * IN + (size_t)(kc + 1) * 32,
                  sF + ((kc + 1) & 1) * (128 * NPAD), tid);

    const float* fb  = fbase  + (kc & 1) * (128 * NPAD);
    const float* b0p = b0base + kc * 32;
    const float* b1p = b1base + kc * 32;
#pragma unroll
    for (int ks = 0; ks < 8; ++ks) {
      int kl = ks * 4 + koff;
      v2f a  = *(const v2f*)(fb  + kl);   // A(m=lanem, k=kl..kl+1)
      v2f b0 = *(const v2f*)(b0p + kl);   // B(k=kl..kl+1, n=lanem)
      v2f b1 = *(const v2f*)(b1p + kl);   // B(k=kl..kl+1, n=16+lanem)
      acc0 = __builtin_amdgcn_wmma_f32_16x16x4_f32(false, a, false, b0,
                                                   (short)0, acc0, false, false);
      acc1 = __builtin_amdgcn_wmma_f32_16x16x4_f32(false, a, false, b1,
                                                   (short)0, acc1, false, false);
    }
  }

  // Epilogue: bias + relu, store padded h. C/D layout: VGPR g, lanes 0-15 ->
  // M=g, lanes 16-31 -> M=g+8, col n = lanem (+16 for tile 1).
  float bias0 = bb0[lanem];
  float bias1 = (16 + lanem < HID) ? bb0[16 + lanem] : 0.0f;
  float* hp = hout + (size_t)(row0 + wv * 16 + hi * 8) * NPAD;
#pragma unroll
  for (int g = 0; g < 8; ++g) {
    float h0 = fmaxf(acc0[g] + bias0, 0.0f);
    float h1 = fmaxf(acc1[g] + bias1, 0.0f);
    hp[(size_t)g * NPAD + lanem]      = h0;
    hp[(size_t)g * NPAD + 16 + lanem] = h1;
  }
}

// ---------------------------------------------------------------------------
// Kernel B: emb = h@bW1+bb1; g = relu(emb . w0[s] + b0[s]); out = g . w1[s]+b1[s]
// All expert tables + bW1 staged into LDS (~300KB, fits 320KB/WGP). Padded
// expert strides (902, 242) decorrelate LDS banks for per-lane subject gather.
// One thread per sample. 128 WGs x 256 threads.
// ---------------------------------------------------------------------------

#define FW0S 902   // padded stride for fW0 experts (900 -> 902)
#define FW1S 242   // padded stride for fW1 experts (240 -> 242)
#define SMEMB_FLOATS (900 + 32 + NSUB * FW0S + NSUB * 32 + NSUB * FW1S + NSUB * ODIM)

__global__ __launch_bounds__(256) void frontend_moe_kernel(
    const float* __restrict__ hws, const int* __restrict__ subject,
    const float* __restrict__ bW1, const float* __restrict__ bb1,
    const float* __restrict__ fW0, const float* __restrict__ fb0,
    const float* __restrict__ fW1, const float* __restrict__ fb1,
    float* __restrict__ out) {
  extern __shared__ float smem[];
  float* sW1  = smem;                                   // 900
  float* sb1  = smem + 900;                             // 32
  float* sFW0 = smem + 932;                             // 64*902
  float* sFb0 = smem + 932 + NSUB * FW0S;               // 64*32
  float* sFW1 = sFb0 + NSUB * 32;                       // 64*242
  float* sFb1 = sFW1 + NSUB * FW1S;                     // 64*8
  const int tid = threadIdx.x;

  for (int i = tid; i < 900; i += 256) sW1[i] = bW1[i];
  if (tid < HID) sb1[tid] = bb1[tid];
  for (int i = tid; i < NSUB * 900; i += 256) {
    int s = i / 900, r = i - s * 900;
    sFW0[s * FW0S + r] = fW0[i];
  }
  for (int i = tid; i < NSUB * HID; i += 256) {
    int s = i / HID, r = i - s * HID;
    sFb0[s * 32 + r] = fb0[i];
  }
  for (int i = tid; i < NSUB * HID * ODIM; i += 256) {
    int s = i / 240, r = i - s * 240;
    sFW1[s * FW1S + r] = fW1[i];
  }
  for (int i = tid; i < NSUB * ODIM; i += 256) sFb1[i] = fb1[i];
  __syncthreads();

  const int b = blockIdx.x * 256 + tid;
  const float* hp = hws + (size_t)b * NPAD;
  float hrow[HID];
#pragma unroll
  for (int j = 0; j < 28; j += 4) {
    v4f t = *(const v4f*)(hp + j);
    hrow[j] = t[0]; hrow[j + 1] = t[1]; hrow[j + 2] = t[2]; hrow[j + 3] = t[3];
  }
  { v2f t = *(const v2f*)(hp + 28); hrow[28] = t[0]; hrow[29] = t[1]; }

  // emb = h @ bW1 + bb1  (uniform addresses -> LDS broadcast reads)
  float emb[HID];
#pragma unroll
  for (int k = 0; k < HID; ++k) emb[k] = sb1[k];
#pragma unroll
  for (int j = 0; j < HID; ++j) {
    float hj = hrow[j];
    const float* wr = sW1 + j * HID;
#pragma unroll
    for (int k = 0; k < HID; k += 2) {
      v2f w = *(const v2f*)(wr + k);
      emb[k]     = fmaf(hj, w[0], emb[k]);
      emb[k + 1] = fmaf(hj, w[1], emb[k + 1]);
    }
  }

  const int s = subject[b];
  // g = relu(emb . w0[s] + b0[s])
  float g[HID];
  const float* fb0p = sFb0 + s * 32;
#pragma unroll
  for (int o = 0; o < HID; ++o) g[o] = fb0p[o];
  const float* w0p = sFW0 + s * FW0S;
#pragma unroll
  for (int k = 0; k < HID; ++k) {
    float ek = emb[k];
    const float* wr = w0p + k * HID;
#pragma unroll
    for (int o = 0; o < HID; o += 2) {
      v2f w = *(const v2f*)(wr + o);
      g[o]     = fmaf(ek, w[0], g[o]);
      g[o + 1] = fmaf(ek, w[1], g[o + 1]);
    }
  }
#pragma unroll
  for (int o = 0; o < HID; ++o) g[o] = fmaxf(g[o], 0.0f);

  // out = g . w1[s] + b1[s]
  float acc[ODIM];
  const float* fb1p = sFb1 + s * ODIM;
#pragma unroll
  for (int o = 0; o < ODIM; ++o) acc[o] = fb1p[o];
  const float* w1p = sFW1 + s * FW1S;
#pragma unroll
  for (int k = 0; k < HID; ++k) {
    float gk = g[k];
    const float* wr = w1p + k * ODIM;
#pragma unroll
    for (int o = 0; o < ODIM; o += 2) {
      v2f w = *(const v2f*)(wr + o);
      acc[o]     = fmaf(gk, w[0], acc[o]);
      acc[o + 1] = fmaf(gk, w[1], acc[o + 1]);
    }
  }
  v4f o0 = {acc[0], acc[1], acc[2], acc[3]};
  v4f o1 = {acc[4], acc[5], acc[6], acc[7]};
  *(v4f*)(out + (size_t)b * ODIM)     = o0;
  *(v4f*)(out + (size_t)b * ODIM + 4) = o1;
}

// ---------------------------------------------------------------------------

extern "C" void kernel_launch(void* const* d_in, const int* in_sizes, int n_in,
                              void* d_out, int out_size, void* d_ws, size_t ws_size,
                              hipStream_t stream) {
  const float* feat    = (const float*)d_in[0];
  const int*   subject = (const int*)d_in[1];
  const float* bW0     = (const float*)d_in[2];
  const float* bb0     = (const float*)d_in[3];
  const float* bW1     = (const float*)d_in[4];
  const float* bb1     = (const float*)d_in[5];
  const float* fW0     = (const float*)d_in[6];
  const float* fb0     = (const float*)d_in[7];
  const float* fW1     = (const float*)d_in[8];
  const float* fb1     = (const float*)d_in[9];
  float* hws = (float*)d_ws;  // needs BATCH*32*4 = 4MB scratch

  size_t smemA = (size_t)(NPAD * IN + 2 * 128 * NPAD) * sizeof(float); // 160KB
  size_t smemB = (size_t)SMEMB_FLOATS * sizeof(float);                 // ~300KB
  (void)hipFuncSetAttribute((const void*)backend_gemm_kernel,
                            hipFuncAttributeMaxDynamicSharedMemorySize, (int)smemA);
  (void)hipFuncSetAttribute((const void*)frontend_moe_kernel,
                            hipFuncAttributeMaxDynamicSharedMemorySize, (int)smemB);

  backend_gemm_kernel<<<BATCH / 128, 256, smemA, stream>>>(feat, bW0, bb0, hws);
  frontend_moe_kernel<<<BATCH / 256, 256, smemB, stream>>>(
      hws, subject, bW1, bb1, fW0, fb0, fW1, fb1, (float*)d_out);
}